// QuantumLayer_9216999817604
// MI455X (gfx1250) — compile-verified
//
#include <hip/hip_runtime.h>

// ---------------------------------------------------------------------------
// 10-qubit batched state-vector simulator for MI455X (gfx1250, wave32, WMMA).
//
// Strategy:
//  * state[1024] complex f32 per sample lives in LDS (padded 17-float rows).
//  * encoding RY gates -> direct Kronecker init (state is separable).
//  * Rot layer -> U(16x16) on wires0-3 via WMMA f32 16x16x4 chains,
//                 V^T(16x16) on wires6-9 via WMMA, W(4x4) on wires4-5 in VALU.
//  * CNOT ring  -> single precomputed-bitwise basis permutation in LDS.
//  * U/W/V^T shared across batch: tiny setup kernel writes them to d_ws.
// ---------------------------------------------------------------------------

typedef float v2f __attribute__((ext_vector_type(2)));
typedef float v8f __attribute__((ext_vector_type(8)));

#define N_QUBITS 10
#define N_LAYERS 6
#define BATCH 4096
#define LAYER_STRIDE 1056   // floats per layer in workspace: Ur,Ui,Br,Bi (256 ea) + Wr,Wi (16 ea)

// padded LDS layout: slice m (middle bits), row t (top nibble), col u (bottom nibble)
#define IDX(m,t,u) ((m)*272 + (t)*17 + (u))

__device__ __forceinline__ float2 cmul(float2 a, float2 b) {
    return make_float2(a.x*b.x - a.y*b.y, a.x*b.y + a.y*b.x);
}

// Rot = RZ(omega) RY(theta) RZ(phi); element g[i][j]
__device__ __forceinline__ float2 rot_elem(float phi, float theta, float omega, int i, int j) {
    float s, c, sp, cp, sm, cm;
    __sincosf(0.5f*theta, &s, &c);
    __sincosf(0.5f*(phi+omega), &sp, &cp);
    __sincosf(0.5f*(phi-omega), &sm, &cm);
    if (i==0 && j==0) return make_float2( c*cp, -c*sp);   // ep*c
    if (i==0)         return make_float2(-s*cm, -s*sm);   // -conj(em)*s
    if (j==0)         return make_float2( s*cm, -s*sm);   // em*s
    return                   make_float2( c*cp,  c*sp);   // conj(ep)*c
}

__device__ __forceinline__ int idx_s(int s) {
    return IDX((s>>4)&3, s>>6, s&15);
}

// inverse of the full CNOT ring permutation on basis labels (wire0 = MSB)
__device__ __forceinline__ int perm_inv(int s) {
    int b[10];
    #pragma unroll
    for (int i = 0; i < 10; ++i) b[i] = (s >> (9-i)) & 1;
    b[0] ^= b[9];                       // undo CNOT(9,0) first
    #pragma unroll
    for (int i = 9; i >= 1; --i) b[i] ^= b[i-1];  // then CNOT(8,9) ... CNOT(0,1)
    int r = 0;
    #pragma unroll
    for (int i = 0; i < 10; ++i) r |= b[i] << (9-i);
    return r;
}

// ---------------------------------------------------------------------------
// Setup: build per-layer gate matrices (shared across batch) into ws.
//   base+0    : Ur[256]   U = kron(g0,g1,g2,g3), row-major [t][t']
//   base+256  : Ui[256]
//   base+512  : Br[256]   B[k][n] = V[n][k],  V = kron(g6,g7,g8,g9)
//   base+768  : Bi[256]
//   base+1024 : Wr[16]    W = kron(g4,g5), row-major [m][m']
//   base+1040 : Wi[16]
// ---------------------------------------------------------------------------
__global__ __launch_bounds__(256) void qc_setup(const float* __restrict__ params,
                                                float* __restrict__ ws) {
    const int l = blockIdx.x;
    const int e = threadIdx.x;
    float* base = ws + l*LAYER_STRIDE;

    {   // U entry e = t*16 + t'
        int t = e >> 4, tp = e & 15;
        float2 acc = make_float2(1.f, 0.f);
        #pragma unroll
        for (int j = 0; j < 4; ++j) {
            const float* p = params + (l*N_QUBITS + j)*3;
            acc = cmul(acc, rot_elem(p[0], p[1], p[2], (t>>(3-j))&1, (tp>>(3-j))&1));
        }
        base[e]       = acc.x;
        base[256 + e] = acc.y;
    }
    {   // B entry e = k*16 + n, value = V[n][k]
        int k = e >> 4, n = e & 15;
        float2 acc = make_float2(1.f, 0.f);
        #pragma unroll
        for (int j = 0; j < 4; ++j) {
            const float* p = params + (l*N_QUBITS + 6 + j)*3;
            acc = cmul(acc, rot_elem(p[0], p[1], p[2], (n>>(3-j))&1, (k>>(3-j))&1));
        }
        base[512 + e] = acc.x;
        base[768 + e] = acc.y;
    }
    if (e < 16) {   // W entry e = m*4 + m'
        int m = e >> 2, mp = e & 3;
        float2 acc = make_float2(1.f, 0.f);
        #pragma unroll
        for (int j = 0; j < 2; ++j) {
            const float* p = params + (l*N_QUBITS + 4 + j)*3;
            acc = cmul(acc, rot_elem(p[0], p[1], p[2], (m>>(1-j))&1, (mp>>(1-j))&1));
        }
        base[1024 + e] = acc.x;
        base[1040 + e] = acc.y;
    }
}

#define WMMA4(a, b, c) \
    __builtin_amdgcn_wmma_f32_16x16x4_f32(false, (a), false, (b), (short)0, (c), false, false)

// ---------------------------------------------------------------------------
// Main: one block per sample, 128 threads = 4 waves; wave w owns slice m = w.
// ---------------------------------------------------------------------------
__global__ __launch_bounds__(128) void qc_main(const float* __restrict__ x,
                                               const float* __restrict__ ws,
                                               float* __restrict__ out) {
    __shared__ float S0r[1088], S0i[1088], S1r[1088], S1i[1088];
    __shared__ float csv[2*N_QUBITS];
    __shared__ float red[128][10];

    const int tid  = threadIdx.x;
    const int lane = tid & 31;
    const int m    = tid >> 5;      // wave id == middle-bit slice
    const int tc   = lane & 15;     // M (or N) index inside the 16x16 tile
    const int h    = lane >> 4;     // lane-half selects K pairs

    // ---- encoding: state = kron_i (cos(x_i/2), sin(x_i/2)) ----
    if (tid < N_QUBITS) {
        float xv = x[blockIdx.x * N_QUBITS + tid];
        float s, c;
        __sincosf(0.5f*xv, &s, &c);
        csv[2*tid]   = c;
        csv[2*tid+1] = s;
    }
    __syncthreads();
    #pragma unroll
    for (int k = 0; k < 8; ++k) {
        int s = tid*8 + k;
        float a = 1.f;
        #pragma unroll
        for (int i = 0; i < 10; ++i) a *= csv[2*i + ((s>>(9-i))&1)];
        S0r[idx_s(s)] = a;
        S0i[idx_s(s)] = 0.f;
    }
    __syncthreads();

    for (int l = 0; l < N_LAYERS; ++l) {
        const float* gUr = ws + l*LAYER_STRIDE;
        const float* gUi = gUr + 256;
        const float* gBr = gUr + 512;
        const float* gBi = gUr + 768;
        const float* gWr = gUr + 1024;
        const float* gWi = gUr + 1040;

        // ---- left stage: T_m = U * S_m   (S0 -> S1) ----
        {
            v2f ar[4], ai[4], ani[4], br[4], bi[4];
            #pragma unroll
            for (int kk = 0; kk < 4; ++kk) {
                int k0 = 4*kk + 2*h;                       // K = k0 (.x) / k0+1 (.y)
                ar[kk].x = gUr[tc*16 + k0]; ar[kk].y = gUr[tc*16 + k0 + 1];
                ai[kk].x = gUi[tc*16 + k0]; ai[kk].y = gUi[tc*16 + k0 + 1];
                ani[kk] = -ai[kk];
                br[kk].x = S0r[IDX(m, k0, tc)]; br[kk].y = S0r[IDX(m, k0+1, tc)];
                bi[kk].x = S0i[IDX(m, k0, tc)]; bi[kk].y = S0i[IDX(m, k0+1, tc)];
            }
            v8f cr = {}, ci = {};
            #pragma unroll
            for (int kk = 0; kk < 4; ++kk) cr = WMMA4(ar[kk],  br[kk], cr);
            #pragma unroll
            for (int kk = 0; kk < 4; ++kk) cr = WMMA4(ani[kk], bi[kk], cr);
            #pragma unroll
            for (int kk = 0; kk < 4; ++kk) ci = WMMA4(ar[kk],  bi[kk], ci);
            #pragma unroll
            for (int kk = 0; kk < 4; ++kk) ci = WMMA4(ai[kk],  br[kk], ci);
            #pragma unroll
            for (int v = 0; v < 8; ++v) {
                int row = v + 8*h;                         // C/D layout: M = vgpr + 8*half
                S1r[IDX(m, row, tc)] = cr[v];
                S1i[IDX(m, row, tc)] = ci[v];
            }
        }
        __syncthreads();

        // ---- right stage: S_m = T_m * V^T   (S1 -> S0) ----
        {
            v2f atr[4], ati[4], brr[4], bri[4], nbri[4];
            #pragma unroll
            for (int kk = 0; kk < 4; ++kk) {
                int k0 = 4*kk + 2*h;
                atr[kk].x = S1r[IDX(m, tc, k0)]; atr[kk].y = S1r[IDX(m, tc, k0+1)];
                ati[kk].x = S1i[IDX(m, tc, k0)]; ati[kk].y = S1i[IDX(m, tc, k0+1)];
                brr[kk].x = gBr[k0*16 + tc];     brr[kk].y = gBr[(k0+1)*16 + tc];
                bri[kk].x = gBi[k0*16 + tc];     bri[kk].y = gBi[(k0+1)*16 + tc];
                nbri[kk] = -bri[kk];
            }
            v8f cr = {}, ci = {};
            #pragma unroll
            for (int kk = 0; kk < 4; ++kk) cr = WMMA4(atr[kk], brr[kk],  cr);
            #pragma unroll
            for (int kk = 0; kk < 4; ++kk) cr = WMMA4(ati[kk], nbri[kk], cr);
            #pragma unroll
            for (int kk = 0; kk < 4; ++kk) ci = WMMA4(atr[kk], bri[kk],  ci);
            #pragma unroll
            for (int kk = 0; kk < 4; ++kk) ci = WMMA4(ati[kk], brr[kk],  ci);
            #pragma unroll
            for (int v = 0; v < 8; ++v) {
                int row = v + 8*h;
                S0r[IDX(m, row, tc)] = cr[v];
                S0i[IDX(m, row, tc)] = ci[v];
            }
        }
        __syncthreads();

        // ---- middle stage: mix the 4 slices with W (4x4)   (S0 -> S1) ----
        #pragma unroll
        for (int kp = 0; kp < 2; ++kp) {
            int p = tid*2 + kp;
            int t = p >> 4, u = p & 15;
            float2 old[4];
            #pragma unroll
            for (int mm = 0; mm < 4; ++mm)
                old[mm] = make_float2(S0r[IDX(mm,t,u)], S0i[IDX(mm,t,u)]);
            #pragma unroll
            for (int mo = 0; mo < 4; ++mo) {
                float2 acc = make_float2(0.f, 0.f);
                #pragma unroll
                for (int mm = 0; mm < 4; ++mm) {
                    float2 w = make_float2(gWr[mo*4+mm], gWi[mo*4+mm]);
                    float2 pr = cmul(w, old[mm]);
                    acc.x += pr.x; acc.y += pr.y;
                }
                S1r[IDX(mo,t,u)] = acc.x;
                S1i[IDX(mo,t,u)] = acc.y;
            }
        }
        __syncthreads();

        // ---- CNOT ring: pure basis permutation   (S1 -> S0) ----
        #pragma unroll
        for (int k = 0; k < 8; ++k) {
            int s  = tid*8 + k;
            int sp = perm_inv(s);
            S0r[idx_s(s)] = S1r[idx_s(sp)];
            S0i[idx_s(s)] = S1i[idx_s(sp)];
        }
        __syncthreads();
    }

    // ---- <Z_q> = sum_s |amp(s)|^2 * (1 - 2*bit_q(s)) ----
    float acc[10];
    #pragma unroll
    for (int q = 0; q < 10; ++q) acc[q] = 0.f;
    #pragma unroll
    for (int k = 0; k < 8; ++k) {
        int s = tid*8 + k;
        int ii = idx_s(s);
        float rr = S0r[ii], im = S0i[ii];
        float p = rr*rr + im*im;
        #pragma unroll
        for (int q = 0; q < 10; ++q) acc[q] += ((s>>(9-q))&1) ? -p : p;
    }
    #pragma unroll
    for (int q = 0; q < 10; ++q) red[tid][q] = acc[q];
    __syncthreads();
    for (int off = 64; off > 0; off >>= 1) {
        if (tid < off) {
            #pragma unroll
            for (int q = 0; q < 10; ++q) red[tid][q] += red[tid+off][q];
        }
        __syncthreads();
    }
    if (tid < N_QUBITS) out[blockIdx.x*N_QUBITS + tid] = red[0][tid];
}

// ---------------------------------------------------------------------------
extern "C" void kernel_launch(void* const* d_in, const int* in_sizes, int n_in,
                              void* d_out, int out_size, void* d_ws, size_t ws_size,
                              hipStream_t stream) {
    const float* x      = (const float*)d_in[0];   // [4096, 10] f32
    const float* params = (const float*)d_in[1];   // [180] f32
    float*       out    = (float*)d_out;           // [4096, 10] f32
    float*       ws     = (float*)d_ws;            // >= 6*1056 floats

    qc_setup<<<N_LAYERS, 256, 0, stream>>>(params, ws);
    qc_main<<<BATCH, 128, 0, stream>>>(x, ws, out);
}